// _TempGCN_65781719105682
// MI455X (gfx1250) — compile-verified
//
#include <hip/hip_runtime.h>
#include <math.h>

// GCN: 2x GCNConv(+self loops, sym norm) + linear head + softplus.
// N=100000, E=3200000, IN_DIM=128, HID=32.

typedef float v2f __attribute__((ext_vector_type(2)));
typedef float v8f __attribute__((ext_vector_type(8)));

#define HID 32
#define TB 256

// ---------------- degree / dinv ----------------

__global__ void k_zero_u32(unsigned* __restrict__ p, int n) {
    int i = blockIdx.x * blockDim.x + threadIdx.x;
    if (i < n) p[i] = 0u;
}

__global__ void k_deg_count(const long long* __restrict__ ei, unsigned* __restrict__ deg, int E) {
    int e = blockIdx.x * blockDim.x + threadIdx.x;
    if (e < E) atomicAdd(&deg[(int)ei[E + e]], 1u);   // dst = ei[1][e]
}

__global__ void k_dinv(const unsigned* __restrict__ deg, float* __restrict__ dinv, int n) {
    int i = blockIdx.x * blockDim.x + threadIdx.x;
    if (i < n) dinv[i] = rsqrtf((float)deg[i] + 1.0f); // +1 self loop
}

// ---------------- WMMA GEMM: C[M x 32] = A[M x K] @ B[K x 32] (fp32) ----------------
// One wave per 16-row block; two 16x16 accumulator tiles cover the 32 output cols.
// V_WMMA_F32_16X16X4_F32 fragment layouts (wave32):
//   A 16x4 : lanes 0-15 -> M=lane, VGPR0=K0, VGPR1=K1 ; lanes 16-31 -> VGPR0=K2, VGPR1=K3
//   B 4x16 : lanes 0-15 -> N=lane, VGPR0=K0 row, VGPR1=K1 row ; lanes 16-31 -> K2,K3 rows
//   C 16x16: VGPR g -> (M=g, N=lane) lanes 0-15 ; (M=g+8, N=lane-16) lanes 16-31

template <int K>
__global__ __launch_bounds__(TB) void k_gemm_wmma(const float* __restrict__ A,
                                                  const float* __restrict__ B,
                                                  float* __restrict__ C, int M) {
    int wave = (blockIdx.x * blockDim.x + threadIdx.x) >> 5;
    int lane = threadIdx.x & 31;
    int row0 = wave * 16;
    if (row0 >= M) return;                       // uniform per wave

    if (row0 + 16 > M) {                         // scalar tail (uniform per wave)
        int r = row0 + lane;
        if (r < M) {
            for (int n = 0; n < HID; ++n) {
                float s = 0.f;
                for (int k = 0; k < K; ++k) s += A[(size_t)r * K + k] * B[(size_t)k * HID + n];
                C[(size_t)r * HID + n] = s;
            }
        }
        return;
    }

    int m  = lane & 15;
    int kh = (lane >> 4) * 2;                    // 0 for lanes 0-15, 2 for lanes 16-31

    v8f acc0 = {0.f, 0.f, 0.f, 0.f, 0.f, 0.f, 0.f, 0.f};
    v8f acc1 = {0.f, 0.f, 0.f, 0.f, 0.f, 0.f, 0.f, 0.f};

    const float* arow = A + (size_t)(row0 + m) * K;
#pragma unroll
    for (int k0 = 0; k0 < K; k0 += 4) {
        v2f a;
        a.x = arow[k0 + kh + 0];
        a.y = arow[k0 + kh + 1];

        const float* bp = B + (size_t)(k0 + kh) * HID;
        v2f b0, b1;
        b0.x = bp[m];            b0.y = bp[HID + m];        // cols 0..15
        b1.x = bp[16 + m];       b1.y = bp[HID + 16 + m];   // cols 16..31

        acc0 = __builtin_amdgcn_wmma_f32_16x16x4_f32(false, a, false, b0,
                                                     (short)0, acc0, false, false);
        acc1 = __builtin_amdgcn_wmma_f32_16x16x4_f32(false, a, false, b1,
                                                     (short)0, acc1, false, false);
    }

    int mbase = row0 + ((lane >> 4) ? 8 : 0);
#pragma unroll
    for (int g = 0; g < 8; ++g) {
        C[(size_t)(mbase + g) * HID + m]      = acc0[g];
        C[(size_t)(mbase + g) * HID + 16 + m] = acc1[g];
    }
}

// ---------------- aggregation ----------------

// agg[i][c] = h[i][c] * dinv[i]^2   (self-loop term, also initializes agg)
__global__ void k_self_init(const float* __restrict__ h, const float* __restrict__ dinv,
                            float* __restrict__ agg, int n) {
    int i = blockIdx.x * blockDim.x + threadIdx.x;
    if (i < n * HID) {
        float di = dinv[i >> 5];
        agg[i] = h[i] * di * di;
    }
}

// one wave per edge, lane = channel: agg[dst][c] += h[src][c] * dinv[src]*dinv[dst]
__global__ void k_edge_agg(const long long* __restrict__ ei, const float* __restrict__ h,
                           const float* __restrict__ dinv, float* __restrict__ agg, int E) {
    int t    = blockIdx.x * blockDim.x + threadIdx.x;
    int e    = t >> 5;
    int lane = t & 31;
    if (e >= E) return;
    int s = (int)ei[e];
    int d = (int)ei[E + e];
    float norm = dinv[s] * dinv[d];
    float v = h[s * HID + lane] * norm;
    atomicAdd(&agg[d * HID + lane], v);
}

// h[i][c] = relu(agg[i][c] + b[c])
__global__ void k_bias_relu(const float* __restrict__ agg, const float* __restrict__ b,
                            float* __restrict__ hout, int n) {
    int i = blockIdx.x * blockDim.x + threadIdx.x;
    if (i < n * HID) hout[i] = fmaxf(agg[i] + b[i & (HID - 1)], 0.f);
}

// out[i] = softplus( sum_c relu(agg[i][c]+b2[c]) * Wout[c] + bout ) + 0.001
__global__ void k_out(const float* __restrict__ agg, const float* __restrict__ b2,
                      const float* __restrict__ Wout, const float* __restrict__ bout,
                      float* __restrict__ out, int n) {
    int t    = blockIdx.x * blockDim.x + threadIdx.x;
    int node = t >> 5;
    int lane = t & 31;
    if (node >= n) return;
    float v = fmaxf(agg[node * HID + lane] + b2[lane], 0.f) * Wout[lane];
#pragma unroll
    for (int off = 16; off > 0; off >>= 1) v += __shfl_xor(v, off, 32);
    if (lane == 0) {
        float tt = v + bout[0];
        float sp = (tt > 20.f) ? tt : log1pf(expf(tt));
        out[node] = sp + 0.001f;
    }
}

// ---------------- launcher ----------------

extern "C" void kernel_launch(void* const* d_in, const int* in_sizes, int n_in,
                              void* d_out, int out_size, void* d_ws, size_t ws_size,
                              hipStream_t stream) {
    const float*     x    = (const float*)d_in[0];
    const long long* ei   = (const long long*)d_in[1];
    const float*     W1   = (const float*)d_in[2];
    const float*     b1   = (const float*)d_in[3];
    const float*     W2   = (const float*)d_in[4];
    const float*     b2   = (const float*)d_in[5];
    const float*     Wout = (const float*)d_in[6];
    const float*     bout = (const float*)d_in[7];
    float*           out  = (float*)d_out;

    const int IN_DIM = 128;
    const int N = in_sizes[0] / IN_DIM;
    const int E = in_sizes[1] / 2;

    // workspace layout: deg[N] u32 | dinv[N] f32 | h[N*32] | h2[N*32] | agg[N*32]
    char*     ws   = (char*)d_ws;
    unsigned* deg  = (unsigned*)ws;
    float*    dinv = (float*)(ws + (size_t)N * 4);
    float*    h    = (float*)(ws + (size_t)N * 8);
    float*    h2   = h + (size_t)N * HID;
    float*    agg  = h2 + (size_t)N * HID;

    int gN   = (N + TB - 1) / TB;
    int gE   = (E + TB - 1) / TB;
    int gNC  = (N * HID + TB - 1) / TB;                 // N*32 = 3.2M threads
    int gE32 = (int)(((long long)E * 32 + TB - 1) / TB);
    int gN32 = (int)(((long long)N * 32 + TB - 1) / TB);
    int waves = (N + 15) / 16;
    int gGemm = (waves * 32 + TB - 1) / TB;

    // degree + dinv
    k_zero_u32 <<<gN, TB, 0, stream>>>(deg, N);
    k_deg_count<<<gE, TB, 0, stream>>>(ei, deg, E);
    k_dinv     <<<gN, TB, 0, stream>>>(deg, dinv, N);

    // layer 1: h = x @ W1 ; agg = scatter + self ; h = relu(agg + b1)
    k_gemm_wmma<128><<<gGemm, TB, 0, stream>>>(x, W1, h, N);
    k_self_init <<<gNC, TB, 0, stream>>>(h, dinv, agg, N);
    k_edge_agg  <<<gE32, TB, 0, stream>>>(ei, h, dinv, agg, E);
    k_bias_relu <<<gNC, TB, 0, stream>>>(agg, b1, h, N);

    // layer 2: h2 = h @ W2 ; agg = scatter + self
    k_gemm_wmma<32><<<gGemm, TB, 0, stream>>>(h, W2, h2, N);
    k_self_init <<<gNC, TB, 0, stream>>>(h2, dinv, agg, N);
    k_edge_agg  <<<gE32, TB, 0, stream>>>(ei, h2, dinv, agg, E);

    // head: fused bias+relu+dot+softplus
    k_out<<<gN32, TB, 0, stream>>>(agg, b2, Wout, bout, out, N);
}